// HierarchicalSoftmaxLoss_11836929868638
// MI455X (gfx1250) — compile-verified
//
#include <hip/hip_runtime.h>
#include <hip/hip_bf16.h>

// HierarchicalSoftmaxLoss for MI455X (gfx1250, wave32)
// B=32768 samples, D=4368 logits/row, 3 levels x 16-wide softmax windows.
// Pass 1: one lane per sample -> per-block partial sums (deterministic).
//         Grid padded to a multiple of 64 blocks; pad blocks write 0.
// Pass 2: single wave32 reduces the padded partials with chained
//         V_WMMA_F32_16X16X4_F32 (A = ones 16x4 => D = column sums of B,
//         accumulated), branch-free b64 loads, 4-step shuffle finish.

typedef float v2f __attribute__((ext_vector_type(2)));
typedef float v8f __attribute__((ext_vector_type(8)));

__device__ __forceinline__ float hsm_level_loss(const float* __restrict__ wptr, int lab) {
    // 16 contiguous floats, 64-byte aligned -> 4x global_load_b128 on one line
    const float4* w = reinterpret_cast<const float4*>(wptr);
    float4 a = w[0], b = w[1], c = w[2], d = w[3];
    float x[16] = {a.x, a.y, a.z, a.w, b.x, b.y, b.z, b.w,
                   c.x, c.y, c.z, c.w, d.x, d.y, d.z, d.w};
    float m = x[0];
#pragma unroll
    for (int i = 1; i < 16; ++i) m = fmaxf(m, x[i]);
    float s = 0.0f;
    float xl = 0.0f;
#pragma unroll
    for (int i = 0; i < 16; ++i) {
        s += __expf(x[i] - m);          // v_exp_f32 hardware transcendental
        xl = (i == lab) ? x[i] : xl;    // cndmask, no dynamic reg indexing
    }
    // nll = -(x_lab - m - log(sum exp(x-m)))
    return m + __logf(s) - xl;
}

__global__ void hsm_partial_kernel(const float* __restrict__ preds,
                                   const int* __restrict__ targets,
                                   float* __restrict__ partial,
                                   int B, int D) {
    __shared__ float wsum[8];
    int b = blockIdx.x * blockDim.x + threadIdx.x;

    float loss = 0.0f;
    if (b < B) {
        int t = targets[b];
        const float* row = preds + (size_t)b * (size_t)D;
        // level 0: start 0,                     label (t>>8)&15
        // level 1: start 16  + (t>>8)*16,       label (t>>4)&15
        // level 2: start 272 + (t>>4)*16,       label  t     &15
        int s1 = 16 + ((t >> 8) << 4);
        int s2 = 272 + ((t >> 4) << 4);
        loss  = hsm_level_loss(row,       (t >> 8) & 15);
        loss += hsm_level_loss(row + s1,  (t >> 4) & 15);
        loss += hsm_level_loss(row + s2,   t       & 15);
    }

    // intra-wave reduction (wave32)
#pragma unroll
    for (int off = 16; off >= 1; off >>= 1)
        loss += __shfl_xor(loss, off, 32);

    int lane = threadIdx.x & 31;
    int wid  = threadIdx.x >> 5;
    if (lane == 0) wsum[wid] = loss;
    __syncthreads();

    if (threadIdx.x == 0) {
        float s = 0.0f;
        int nw = blockDim.x >> 5;
#pragma unroll
        for (int i = 0; i < 8; ++i)
            if (i < nw) s += wsum[i];
        partial[blockIdx.x] = s;   // pad blocks (no samples) write exactly 0
    }
}

// Single wave32: nPartials is a multiple of 64 -> branch-free b64 loads,
// one chained f32 WMMA per 64 partials, shuffle finish, lane 0 writes mean.
__global__ void hsm_final_kernel(const float* __restrict__ partial,
                                 float* __restrict__ out,
                                 int nPartials, float invB) {
    int lane = threadIdx.x;  // 0..31, EXEC all ones for WMMA

    v2f A;
    A.x = 1.0f; A.y = 1.0f;          // all-ones 16x4 A (B-layout-invariant)
    v8f C = {};

    const float2* p2 = reinterpret_cast<const float2*>(partial);
    for (int base = 0; base < (nPartials >> 1); base += 32) {  // wave-uniform
        float2 bv = p2[base + lane];                           // global_load_b64
        v2f Bv;
        Bv.x = bv.x;
        Bv.y = bv.y;
        // D[m][n] = sum_k A[m][k]*B[k][n] + C[m][n]
        C = __builtin_amdgcn_wmma_f32_16x16x4_f32(
                /*neg_a=*/false, A, /*neg_b=*/false, Bv,
                /*c_mod=*/(short)0, C, /*reuse_a=*/false, /*reuse_b=*/false);
    }

    // VGPR0 of D: lanes 0-15 hold row-0 column sums (rows identical, A = ones).
    float v = C[0];
#pragma unroll
    for (int off = 8; off >= 1; off >>= 1)
        v += __shfl_xor(v, off, 32);
    if (lane == 0) out[0] = v * invB;
}

extern "C" void kernel_launch(void* const* d_in, const int* in_sizes, int n_in,
                              void* d_out, int out_size, void* d_ws, size_t ws_size,
                              hipStream_t stream) {
    const float* preds   = (const float*)d_in[0];
    const int*   targets = (const int*)d_in[1];
    float*       out     = (float*)d_out;
    float*       partial = (float*)d_ws;

    int B = in_sizes[1];                 // 32768
    int D = in_sizes[0] / B;             // 4368

    const int threads = 256;
    int blocks = (B + threads - 1) / threads;   // 128 for B=32768
    int blocksPadded = (blocks + 63) & ~63;     // multiple of 64 -> branch-free pass 2

    hsm_partial_kernel<<<blocksPadded, threads, 0, stream>>>(preds, targets, partial, B, D);
    hsm_final_kernel<<<1, 32, 0, stream>>>(partial, out, blocksPadded, 1.0f / (float)B);
}